// CoraNodeClassification_61100204753130
// MI455X (gfx1250) — compile-verified
//
#include <hip/hip_runtime.h>
#include <hip/hip_bf16.h>
#include <math.h>

#define N_NODES 200000
#define N_EDGES 6400000
#define D_IN    128
#define D_HID   16
#define D_OUT   172

typedef __attribute__((ext_vector_type(2))) float v2f;
typedef __attribute__((ext_vector_type(8))) float v8f;

// ---------------------------------------------------------------------------
// Degree / normalization
// ---------------------------------------------------------------------------
__global__ void deg_init(float* __restrict__ dinv) {
    int i = blockIdx.x * blockDim.x + threadIdx.x;
    if (i < N_NODES) dinv[i] = 1.0f;   // self-loop contributes 1 to in-degree
}

__global__ void deg_count(const int* __restrict__ dst, float* __restrict__ dinv) {
    int e = blockIdx.x * blockDim.x + threadIdx.x;
    if (e < N_EDGES) atomicAdd(&dinv[dst[e]], 1.0f);
}

__global__ void deg_rsqrt(float* __restrict__ dinv) {
    int i = blockIdx.x * blockDim.x + threadIdx.x;
    if (i < N_NODES) dinv[i] = rsqrtf(dinv[i]);
}

// ---------------------------------------------------------------------------
// GEMM1: lin1[200000,16] = x[200000,128] @ W1[128,16]   (fp32 WMMA 16x16x4)
// One wave per 16-row tile (N = 16 = D_HID exactly; 200000/16 = 12500 tiles).
// ---------------------------------------------------------------------------
__global__ void gemm1_wmma(const float* __restrict__ x,
                           const float* __restrict__ W1,
                           float* __restrict__ lin1) {
    const int wid  = (blockIdx.x * blockDim.x + threadIdx.x) >> 5;
    const int lane = threadIdx.x & 31;
    if (wid >= N_NODES / 16) return;               // wave-uniform guard
    const int half = lane >> 4;                    // 0: K pair {0,1}, 1: {2,3}
    const int l    = lane & 15;
    const int m0   = wid * 16;
    const size_t rowA = (size_t)(m0 + l) * D_IN;

    v8f c = {};
    for (int kc = 0; kc < D_IN; kc += 4) {
        const int k = kc + 2 * half;
        // A 16x4 f32 layout: lanes 0-15 hold (M=l, K=kc..kc+1), lanes 16-31 K=kc+2..kc+3
        const float2 av = *(const float2*)(x + rowA + k);
        v2f a; a.x = av.x; a.y = av.y;
        // B 4x16 f32 layout: lane l holds column N=l; same K split by half-wave
        v2f b; b.x = W1[k * D_HID + l]; b.y = W1[(k + 1) * D_HID + l];
        c = __builtin_amdgcn_wmma_f32_16x16x4_f32(false, a, false, b,
                                                  (short)0, c, false, false);
    }
    // C/D layout: lane<16 -> (M=v, N=l), lane>=16 -> (M=v+8, N=l)
    for (int v = 0; v < 8; ++v) {
        const int row = m0 + v + 8 * half;
        lin1[(size_t)row * D_HID + l] = c[v];
    }
}

// ---------------------------------------------------------------------------
// h init (bias + self-loop term), scatter-add over edges, ReLU
// ---------------------------------------------------------------------------
__global__ void init_h(const float* __restrict__ lin1, const float* __restrict__ dinv,
                       const float* __restrict__ b1, float* __restrict__ h) {
    int i = blockIdx.x * blockDim.x + threadIdx.x;
    if (i < N_NODES * D_HID) {
        const int node = i >> 4, f = i & 15;
        const float di = dinv[node];
        h[i] = b1[f] + lin1[i] * di * di;          // self-loop norm = 1/deg
    }
}

__global__ void scatter1(const float* __restrict__ lin1, const float* __restrict__ dinv,
                         const int* __restrict__ src, const int* __restrict__ dst,
                         float* __restrict__ h) {
    unsigned t = blockIdx.x * blockDim.x + threadIdx.x;   // edge*16 + feat
    if (t >= (unsigned)N_EDGES * D_HID) return;
    const int e = (int)(t >> 4);
    const int f = (int)(t & 15);
    const int s = src[e], d = dst[e];
    const float w = dinv[s] * dinv[d];
    atomicAdd(&h[(size_t)d * D_HID + f], lin1[(size_t)s * D_HID + f] * w);
}

__global__ void relu_k(float* __restrict__ h) {
    int i = blockIdx.x * blockDim.x + threadIdx.x;
    if (i < N_NODES * D_HID) h[i] = fmaxf(h[i], 0.0f);
}

// ---------------------------------------------------------------------------
// GEMM2: lin2[200000,172] = h[200000,16] @ W2[16,172]   (fp32 WMMA 16x16x4)
// One wave per 16x16 output tile; 11 column tiles cover N=172 (last padded).
// ---------------------------------------------------------------------------
__global__ void gemm2_wmma(const float* __restrict__ h,
                           const float* __restrict__ W2,
                           float* __restrict__ lin2) {
    const int NTILES = (N_NODES / 16) * 11;        // 137500
    const int wid  = (blockIdx.x * blockDim.x + threadIdx.x) >> 5;
    const int lane = threadIdx.x & 31;
    if (wid >= NTILES) return;                     // wave-uniform guard
    const int tm = wid / 11, tn = wid % 11;
    const int half = lane >> 4, l = lane & 15;
    const int m0 = tm * 16;
    const int n  = tn * 16 + l;
    const bool nok = (n < D_OUT);

    v8f c = {};
    for (int kc = 0; kc < D_HID; kc += 4) {
        const int k = kc + 2 * half;
        const size_t ra = (size_t)(m0 + l) * D_HID;
        v2f a; a.x = h[ra + k]; a.y = h[ra + k + 1];
        v2f b;
        b.x = nok ? W2[k * D_OUT + n] : 0.0f;
        b.y = nok ? W2[(k + 1) * D_OUT + n] : 0.0f;
        c = __builtin_amdgcn_wmma_f32_16x16x4_f32(false, a, false, b,
                                                  (short)0, c, false, false);
    }
    if (nok) {
        for (int v = 0; v < 8; ++v) {
            const int row = m0 + v + 8 * half;
            lin2[(size_t)row * D_OUT + n] = c[v];
        }
    }
}

// ---------------------------------------------------------------------------
// out init (bias + self-loop), wave-per-edge scatter, in-place log_softmax
// ---------------------------------------------------------------------------
__global__ void init_out(const float* __restrict__ lin2, const float* __restrict__ dinv,
                         const float* __restrict__ b2, float* __restrict__ out) {
    int i = blockIdx.x * blockDim.x + threadIdx.x;
    if (i < N_NODES * D_OUT) {
        const int node = i / D_OUT, cc = i % D_OUT;
        const float di = dinv[node];
        out[i] = b2[cc] + lin2[i] * di * di;
    }
}

__global__ void scatter2(const float* __restrict__ lin2, const float* __restrict__ dinv,
                         const int* __restrict__ src, const int* __restrict__ dst,
                         float* __restrict__ out) {
    const int wid  = (blockIdx.x * blockDim.x + threadIdx.x) >> 5;
    const int lane = threadIdx.x & 31;
    if (wid >= N_EDGES) return;
    const int s = src[wid], d = dst[wid];
    const float w = dinv[s] * dinv[d];
    const float* ps = lin2 + (size_t)s * D_OUT;
    float* pd = out + (size_t)d * D_OUT;
    for (int cc = lane; cc < D_OUT; cc += 32)
        atomicAdd(&pd[cc], ps[cc] * w);
}

__global__ void logsoftmax_k(float* __restrict__ out) {
    const int wid  = (blockIdx.x * blockDim.x + threadIdx.x) >> 5;
    const int lane = threadIdx.x & 31;
    if (wid >= N_NODES) return;
    float* row = out + (size_t)wid * D_OUT;
    float m = -INFINITY;
    for (int cc = lane; cc < D_OUT; cc += 32) m = fmaxf(m, row[cc]);
    for (int off = 16; off; off >>= 1) m = fmaxf(m, __shfl_xor(m, off, 32));
    float s = 0.0f;
    for (int cc = lane; cc < D_OUT; cc += 32) s += expf(row[cc] - m);
    for (int off = 16; off; off >>= 1) s += __shfl_xor(s, off, 32);
    const float lse = m + logf(s);
    for (int cc = lane; cc < D_OUT; cc += 32) row[cc] -= lse;
}

// ---------------------------------------------------------------------------
extern "C" void kernel_launch(void* const* d_in, const int* in_sizes, int n_in,
                              void* d_out, int out_size, void* d_ws, size_t ws_size,
                              hipStream_t stream) {
    const float* x  = (const float*)d_in[0];
    const int*   ei = (const int*)d_in[1];      // [2, E] flat: src then dst
    const float* W1 = (const float*)d_in[2];
    const float* b1 = (const float*)d_in[3];
    const float* W2 = (const float*)d_in[4];
    const float* b2 = (const float*)d_in[5];
    const int* src = ei;
    const int* dst = ei + N_EDGES;
    float* out = (float*)d_out;

    // Workspace layout (floats): dinv | lin1 | h | lin2   (~164 MB total)
    float* ws   = (float*)d_ws;
    float* dinv = ws;                                  // 200000
    float* lin1 = ws + 204800;                         // 200000*16
    float* hbuf = ws + 204800 + N_NODES * D_HID;       // 200000*16
    float* lin2 = ws + 204800 + 2 * N_NODES * D_HID;   // 200000*172

    const int B = 256;
    deg_init <<<(N_NODES + B - 1) / B, B, 0, stream>>>(dinv);
    deg_count<<<(N_EDGES + B - 1) / B, B, 0, stream>>>(dst, dinv);
    deg_rsqrt<<<(N_NODES + B - 1) / B, B, 0, stream>>>(dinv);

    // Layer 1
    gemm1_wmma<<<((N_NODES / 16) + 7) / 8, B, 0, stream>>>(x, W1, lin1);
    init_h   <<<(N_NODES * D_HID + B - 1) / B, B, 0, stream>>>(lin1, dinv, b1, hbuf);
    scatter1 <<<(N_EDGES * D_HID / B), B, 0, stream>>>(lin1, dinv, src, dst, hbuf);
    relu_k   <<<(N_NODES * D_HID + B - 1) / B, B, 0, stream>>>(hbuf);

    // Layer 2
    gemm2_wmma<<<(((N_NODES / 16) * 11) + 7) / 8, B, 0, stream>>>(hbuf, W2, lin2);
    init_out  <<<(N_NODES * D_OUT + B - 1) / B, B, 0, stream>>>(lin2, dinv, b2, out);
    scatter2  <<<(N_EDGES * 32 + B - 1) / B, B, 0, stream>>>(lin2, dinv, src, dst, out);
    logsoftmax_k<<<(N_NODES + 7) / 8, B, 0, stream>>>(out);
}